// NeuralResidentialHMM_21680994910766
// MI455X (gfx1250) — compile-verified
//
#include <hip/hip_runtime.h>
#include <math.h>

// Problem constants (from reference): B=64, T=256, S=96, D=256, V=1024, R=16
#define Bsz  64
#define Tlen 256
#define Sst  96
#define Dd   256
#define Vv   1024
#define Rr   16

typedef __attribute__((ext_vector_type(2))) float v2f;
typedef __attribute__((ext_vector_type(8))) float v8f;

static __device__ __forceinline__ v8f v8f_zero() {
  v8f v;
#pragma unroll
  for (int i = 0; i < 8; ++i) v[i] = 0.0f;
  return v;
}

// Reductions within 16-lane halves (masks <= 8 stay inside a half)
static __device__ __forceinline__ float hmax16(float v) {
#pragma unroll
  for (int m = 8; m >= 1; m >>= 1) v = fmaxf(v, __shfl_xor(v, m, 32));
  return v;
}
static __device__ __forceinline__ float hsum16(float v) {
#pragma unroll
  for (int m = 8; m >= 1; m >>= 1) v += __shfl_xor(v, m, 32);
  return v;
}
// Full-wave32 reductions
static __device__ __forceinline__ float wsum32(float v) {
#pragma unroll
  for (int m = 16; m >= 1; m >>= 1) v += __shfl_xor(v, m, 32);
  return v;
}
static __device__ __forceinline__ float wmax32(float v) {
#pragma unroll
  for (int m = 16; m >= 1; m >>= 1) v = fmaxf(v, __shfl_xor(v, m, 32));
  return v;
}

// ---------------------------------------------------------------------------
// Kernel 1: emis[s, v] = softmax_v(state_emb[s,:] . obs_emb[v,:] + bias[v])
// grid = 6 (16 states per block), block = 256 (8 waves).
// Wave w computes 8 N-tiles of 16 vocab columns; K = 256 in chunks of 4 via
// V_WMMA_F32_16X16X4_F32 (full f32, matches reference precision).
// f32 A 16x4 lane layout: VGPR0 = A[m][k + 2*hi], VGPR1 = A[m][k + 2*hi + 1]
// f32 B 4x16 lane layout: VGPR0 = B[k + 2*hi][n],  VGPR1 = B[k + 2*hi + 1][n]
// -> both fragments are contiguous float2 loads from row-major f32 memory.
// C/D: VGPR g holds row m = g + 8*hi, column n = lane & 15.
// ---------------------------------------------------------------------------
__global__ __launch_bounds__(256)
void hmm_emis_kernel(const float* __restrict__ state_emb,
                     const float* __restrict__ obs_emb,
                     const float* __restrict__ bias,
                     float* __restrict__ emis) {
  __shared__ float red[8 * 16];
  __shared__ float rowmax_s[16];
  __shared__ float rowinv_s[16];

  const int tid = threadIdx.x;
  const int w  = tid >> 5;
  const int l  = tid & 31;
  const int ln = l & 15;
  const int hi = l >> 4;
  const int hi2 = hi * 2;
  const int s0 = blockIdx.x * 16;

  const float* Arow = state_emb + (s0 + ln) * Dd + hi2;
  const float* Brow[8];
  int vcol[8];
#pragma unroll
  for (int i = 0; i < 8; ++i) {
    vcol[i] = (w * 8 + i) * 16 + ln;
    Brow[i] = obs_emb + vcol[i] * Dd + hi2;
  }

  v8f acc[8];
#pragma unroll
  for (int i = 0; i < 8; ++i) acc[i] = v8f_zero();

  for (int k = 0; k < Dd; k += 4) {
    v2f a = *(const v2f*)(Arow + k);   // shared across the 8 N-tiles
#pragma unroll
    for (int i = 0; i < 8; ++i) {
      v2f b = *(const v2f*)(Brow[i] + k);
      acc[i] = __builtin_amdgcn_wmma_f32_16x16x4_f32(
          false, a, false, b, (short)0, acc[i], false, false);
    }
  }

  // add bias[v] (column-indexed: same for every row g in a lane)
#pragma unroll
  for (int i = 0; i < 8; ++i) {
    float bv = bias[vcol[i]];
#pragma unroll
    for (int g = 0; g < 8; ++g) acc[i][g] += bv;
  }

  // ---- row max over V (in-register tiles -> half-wave shfl -> LDS across waves)
  float gred[8];
#pragma unroll
  for (int g = 0; g < 8; ++g) {
    float m = -INFINITY;
#pragma unroll
    for (int i = 0; i < 8; ++i) m = fmaxf(m, acc[i][g]);
    gred[g] = hmax16(m);
  }
  if (ln == 0) {
#pragma unroll
    for (int g = 0; g < 8; ++g) red[w * 16 + g + 8 * hi] = gred[g];
  }
  __syncthreads();
  if (tid < 16) {
    float m = -INFINITY;
    for (int w2 = 0; w2 < 8; ++w2) m = fmaxf(m, red[w2 * 16 + tid]);
    rowmax_s[tid] = m;
  }
  __syncthreads();
  float rmx[8];
#pragma unroll
  for (int g = 0; g < 8; ++g) rmx[g] = rowmax_s[g + 8 * hi];

  // ---- row sum of exp
#pragma unroll
  for (int g = 0; g < 8; ++g) {
    float s = 0.0f;
#pragma unroll
    for (int i = 0; i < 8; ++i) s += expf(acc[i][g] - rmx[g]);
    gred[g] = hsum16(s);
  }
  if (ln == 0) {
#pragma unroll
    for (int g = 0; g < 8; ++g) red[w * 16 + g + 8 * hi] = gred[g];
  }
  __syncthreads();
  if (tid < 16) {
    float s = 0.0f;
    for (int w2 = 0; w2 < 8; ++w2) s += red[w2 * 16 + tid];
    rowinv_s[tid] = 1.0f / s;
  }
  __syncthreads();
  float rin[8];
#pragma unroll
  for (int g = 0; g < 8; ++g) rin[g] = rowinv_s[g + 8 * hi];

  // ---- write normalized probabilities
#pragma unroll
  for (int i = 0; i < 8; ++i) {
#pragma unroll
    for (int g = 0; g < 8; ++g) {
      emis[(s0 + g + 8 * hi) * Vv + vcol[i]] = expf(acc[i][g] - rmx[g]) * rin[g];
    }
  }
}

// ---------------------------------------------------------------------------
// Kernel 2: small softmaxes.
// blocks 0..95: trans row i (softmax over 95 off-diagonal targets, scattered
//               to trans[i][0..95] with zero diagonal) + res row i (16-wide).
// block 96:     pi = softmax(unnorm_init) over 96.
// One wave32 per block.
// ---------------------------------------------------------------------------
__global__ __launch_bounds__(32)
void hmm_prep_kernel(const float* __restrict__ utrans,
                     const float* __restrict__ ures,
                     const float* __restrict__ uinit,
                     float* __restrict__ trans,
                     float* __restrict__ res,
                     float* __restrict__ pi) {
  const int i = blockIdx.x;
  const int l = threadIdx.x;
  if (i < Sst) {
    // ---- trans row i: softmax over 95 elements
    const float* row = utrans + i * (Sst - 1);
    float m = -INFINITY;
    for (int j = l; j < Sst - 1; j += 32) m = fmaxf(m, row[j]);
    m = wmax32(m);
    float s = 0.0f;
    for (int j = l; j < Sst - 1; j += 32) s += expf(row[j] - m);
    s = wsum32(s);
    const float inv = 1.0f / s;
    // scatter: col c -> 0 on diagonal; raw[c] for c<i; raw[c-1] for c>i
    for (int c = l; c < Sst; c += 32) {
      float v = 0.0f;
      if (c != i) v = expf(row[(c < i) ? c : (c - 1)] - m) * inv;
      trans[i * Sst + c] = v;
    }
    // ---- res row i: softmax over 16 (lanes 16..31 padded out)
    float u = (l < Rr) ? ures[i * Rr + l] : -INFINITY;
    float rm = wmax32(u);
    float e = (l < Rr) ? expf(u - rm) : 0.0f;
    float rs = wsum32(e);
    if (l < Rr) res[i * Rr + l] = e / rs;
  } else {
    // ---- pi: softmax over 96
    float m = -INFINITY;
    for (int j = l; j < Sst; j += 32) m = fmaxf(m, uinit[j]);
    m = wmax32(m);
    float s = 0.0f;
    for (int j = l; j < Sst; j += 32) s += expf(uinit[j] - m);
    s = wsum32(s);
    const float inv = 1.0f / s;
    for (int j = l; j < Sst; j += 32) pi[j] = expf(uinit[j] - m) * inv;
  }
}

// ---------------------------------------------------------------------------
// Kernel 3: the forward scan. grid = 4, block = 512 (16 waves).
// Wave w owns batch bg = blockIdx*16 + w; lane l owns states {l, l+32, l+64},
// keeping alpha[3][16] and res[3][16] fully in VGPRs (occupancy irrelevant:
// 4 blocks total; the scan is latency-bound, so registers > LDS traffic).
//
// Per step (critical path, 255 serial iterations):
//   1) lanes publish alpha[:,0] to LDS panel a0[16 batches][98] (pad 98 so
//      the (k,k+1) A-fragment pair is 8B-aligned -> single ds_load_b64)
//   2) emission gathers + token load are issued BEFORE the barrier so their
//      L2 latency overlaps the matmul
//   3) waves 0..5 compute one 16x16 tile of enter = a0 @ trans via 24
//      V_WMMA_F32_16X16X4_F32 (K=96). trans is pre-swizzled ONCE into a
//      fragment-ordered buffer transW[kk][half][n][2] so each lane's B
//      fragment is one contiguous ds_load_b64 -> 2 DS ops + 1 WMMA per chunk
//   4) all waves: new = e_t * (alpha_shift + enter*res), full-wave shfl
//      reduction for z, normalize in registers, ll += logf(z)
// 2 barriers per step; emis (384 KB) is L2-resident; next token's column is
// prefetched into near cache (global_prefetch_b8).
// ---------------------------------------------------------------------------
__global__ __launch_bounds__(512)
void hmm_scan_kernel(const int* __restrict__ x,
                     const float* __restrict__ emis,
                     const float* __restrict__ transg,
                     const float* __restrict__ resg,
                     const float* __restrict__ pig,
                     float* __restrict__ out) {
  // transW[kk][h][n][p] = trans[4*kk + 2*h + p][n]; kk=0..23, h=0..1, n=0..95
  __shared__ float transW[24 * 2 * Sst * 2];  // 36 KB, fragment-ordered
  __shared__ float a0_lds[16 * 98];           // [batch m][state k], pad 98
  __shared__ float enter_lds[16 * 97];        // [batch m][state n], pad 97

  const int tid = threadIdx.x;             // 0..511
  const int w   = tid >> 5;                // wave id == local batch 0..15
  const int l   = tid & 31;
  const int bg  = blockIdx.x * 16 + w;     // global batch
  const int ln  = l & 15;
  const int hi  = l >> 4;
  const int hi2 = hi * 2;                  // K sub-offset for f32 WMMA frags

  // ---- pre-swizzle trans into WMMA B-fragment order (once)
  for (int idx = tid; idx < 24 * 2 * Sst; idx += 512) {
    const int kk  = idx / (2 * Sst);
    const int h   = (idx / Sst) & 1;
    const int n   = idx % Sst;
    const int row = kk * 4 + h * 2;
    transW[idx * 2 + 0] = transg[row * Sst + n];
    transW[idx * 2 + 1] = transg[(row + 1) * Sst + n];
  }

  // per-thread residual distribution + pi (registers)
  float resr[3][Rr];
  float pi3[3];
#pragma unroll
  for (int j = 0; j < 3; ++j) {
    const int s = l + 32 * j;
    pi3[j] = pig[s];
#pragma unroll
    for (int r = 0; r < Rr; ++r) resr[j][r] = resg[s * Rr + r];
  }
  __syncthreads();

  // ---- t = 0: alpha0 = e0 * pi * res, normalize, ll = log z0
  float alpha[3][Rr];
  float ll;
  {
    const int xt = x[bg * Tlen + 0];
    float zp = 0.0f;
#pragma unroll
    for (int j = 0; j < 3; ++j) {
      const int s = l + 32 * j;
      const float ep = emis[s * Vv + xt] * pi3[j];
#pragma unroll
      for (int r = 0; r < Rr; ++r) {
        const float v = ep * resr[j][r];
        alpha[j][r] = v;
        zp += v;
      }
    }
    zp = wsum32(zp);
    ll = logf(zp);
    const float rz = 1.0f / zp;
#pragma unroll
    for (int j = 0; j < 3; ++j)
#pragma unroll
      for (int r = 0; r < Rr; ++r) alpha[j][r] *= rz;
  }

  // WMMA operand base addresses (waves 0..5 compute N-tile w)
  const float* Abase = a0_lds + ln * 98 + hi2;                // A[m=ln][k+hi2(+1)]
  const float* Bbase = transW + (hi * Sst + w * 16 + ln) * 2; // B frag pair

  for (int t = 1; t < Tlen; ++t) {
    // 1) stage alpha[:, :, 0]
#pragma unroll
    for (int j = 0; j < 3; ++j) a0_lds[w * 98 + (l + 32 * j)] = alpha[j][0];

    // 2) issue this step's global loads early: overlap L2 latency with WMMA
    const int xt = x[bg * Tlen + t];
    float e3[3];
#pragma unroll
    for (int j = 0; j < 3; ++j) e3[j] = emis[(l + 32 * j) * Vv + xt];

    __syncthreads();

    // 3) enter = a0 @ trans (f32 WMMA, K=96 in 24 chunks of 4)
    if (w < 6) {  // wave-uniform branch: EXEC stays all-ones for WMMA
      v8f acc = v8f_zero();
#pragma unroll
      for (int kk = 0; kk < 24; ++kk) {
        v2f a = *(const v2f*)(Abase + kk * 4);          // ds_load_b64
        v2f b = *(const v2f*)(Bbase + kk * (2 * Sst * 2)); // ds_load_b64
        acc = __builtin_amdgcn_wmma_f32_16x16x4_f32(
            false, a, false, b, (short)0, acc, false, false);
      }
      // D tile: VGPR g -> batch row m = g + 8*hi, col n = w*16 + ln
#pragma unroll
      for (int g = 0; g < 8; ++g)
        enter_lds[(g + 8 * hi) * 97 + w * 16 + ln] = acc[g];
    }
    __syncthreads();

    // 4) elementwise residual-shift update + normalization
    float zp = 0.0f;
    float nv[3][Rr];
#pragma unroll
    for (int j = 0; j < 3; ++j) {
      const int s = l + 32 * j;
      const float e  = e3[j];
      const float en = enter_lds[w * 97 + s];
#pragma unroll
      for (int r = 0; r < Rr - 1; ++r) {
        const float v = e * (alpha[j][r + 1] + en * resr[j][r]);
        nv[j][r] = v;
        zp += v;
      }
      const float v = e * (en * resr[j][Rr - 1]);
      nv[j][Rr - 1] = v;
      zp += v;
    }
    zp = wsum32(zp);
    ll += logf(zp);
    const float rz = 1.0f / zp;
#pragma unroll
    for (int j = 0; j < 3; ++j)
#pragma unroll
      for (int r = 0; r < Rr; ++r) alpha[j][r] = nv[j][r] * rz;

    // prefetch next token's emission column into near cache
    if (t + 1 < Tlen) {
      const int xn = x[bg * Tlen + t + 1];
#pragma unroll
      for (int j = 0; j < 3; ++j)
        __builtin_prefetch(emis + (l + 32 * j) * Vv + xn, 0, 0);
    }
  }

  if (l == 0) out[bg] = ll;
}

// ---------------------------------------------------------------------------
// Host launcher
// ---------------------------------------------------------------------------
extern "C" void kernel_launch(void* const* d_in, const int* in_sizes, int n_in,
                              void* d_out, int out_size, void* d_ws, size_t ws_size,
                              hipStream_t stream) {
  (void)in_sizes; (void)n_in; (void)out_size; (void)ws_size;
  const int*   x    = (const int*)d_in[0];     // [B, T]
  const float* semb = (const float*)d_in[1];   // [S, D]
  const float* oemb = (const float*)d_in[2];   // [V, D]
  const float* bias = (const float*)d_in[3];   // [V]
  const float* utr  = (const float*)d_in[4];   // [S, S-1]
  const float* ure  = (const float*)d_in[5];   // [S, R]
  const float* uin  = (const float*)d_in[6];   // [S]
  float* out = (float*)d_out;                  // [B]

  float* ws    = (float*)d_ws;
  float* emis  = ws;                                   // 96*1024
  float* trans = emis + Sst * Vv;                      // 96*96
  float* res   = trans + Sst * Sst;                    // 96*16
  float* pi    = res + Sst * Rr;                       // 96

  hipLaunchKernelGGL(hmm_emis_kernel, dim3(Sst / 16), dim3(256), 0, stream,
                     semb, oemb, bias, emis);
  hipLaunchKernelGGL(hmm_prep_kernel, dim3(Sst + 1), dim3(32), 0, stream,
                     utr, ure, uin, trans, res, pi);
  hipLaunchKernelGGL(hmm_scan_kernel, dim3(Bsz / 16), dim3(512), 0, stream,
                     x, emis, trans, res, pi, out);
}